// LowFrequencyPath_48198122996216
// MI455X (gfx1250) — compile-verified
//
#include <hip/hip_runtime.h>
#include <cmath>

typedef __attribute__((ext_vector_type(2))) float v2f;
typedef __attribute__((ext_vector_type(8))) float v8f;

#define N_SEG_  40000
#define N_REG_  1000
#define N_EDGE_ 640000
#define HID_    256

// GEMM tiling
#define BM_ 128          // rows per block (8 waves x 16)
#define BN_ 64           // cols per block (4 wmma tiles per wave)
#define KC_ 32           // K chunk staged in LDS
#define ALD_ 36          // LDS stride (floats) for A rows   (16B aligned, conflict-spread)
#define BLD_ 36          // LDS stride (floats) for B columns

// ---------------------------------------------------------------------------
// WMMA f32 GEMM:  C[M,N] (+)= A[M,K] * B[K,N]   (row-major)
// LDS double-buffered, software-pipelined; inner loop = ds_load_b64 + v_wmma.
// ---------------------------------------------------------------------------
template <bool ACC>
__global__ __launch_bounds__(256) void wmma_gemm_f32(
    const float* __restrict__ A, const float* __restrict__ B,
    float* __restrict__ C, int M, int K, int lda, int ldb, int ldc)
{
    __shared__ float As[2][BM_ * ALD_];   // [row][k]  per chunk
    __shared__ float Bs[2][BN_ * BLD_];   // [col][k]  (transposed) per chunk

    const int tid  = threadIdx.x;
    const int lane = tid & 31;
    const int wave = tid >> 5;                    // uniform per wave
    const int rowBase = blockIdx.x * BM_;
    const int nBase   = blockIdx.y * BN_;
    const int mTile   = blockIdx.x * 8 + wave;    // 16-row tile of this wave

    // cooperative-load mapping
    const int aRow = tid >> 1;                    // 0..127
    const int aCol = (tid & 1) * 16;              // 0 or 16 (4 float4 each)

    // WMMA fragment lane mapping (CDNA5 ISA §7.12.2)
    const int mRow  = lane & 15;
    const int kHalf = (lane >> 4) * 2;            // K 0,1 vs 2,3
    const int nCol  = lane & 15;
    const int mHalf = (lane >> 4) * 8;            // C rows +0..7 vs +8..15

    const bool rowValid = (mTile * 16 < M);       // uniform per wave

    v8f c[4];
#pragma unroll
    for (int j = 0; j < 4; ++j) {
        if (ACC && rowValid) {
#pragma unroll
            for (int r = 0; r < 8; ++r)
                c[j][r] = C[(mTile * 16 + mHalf + r) * ldc + nBase + j * 16 + nCol];
        } else {
            c[j] = (v8f){0.f, 0.f, 0.f, 0.f, 0.f, 0.f, 0.f, 0.f};
        }
    }

    float4 aReg[4];
    float4 bReg[2];

    auto loadChunk = [&](int k0) {
        const int gr = rowBase + aRow;
        const float* __restrict__ ap = A + gr * lda + k0 + aCol;
#pragma unroll
        for (int i = 0; i < 4; ++i) {
            float4 v = {0.f, 0.f, 0.f, 0.f};
            const int gc = k0 + aCol + i * 4;
            if (gr < M) {
                if (gc + 3 < K) {
                    v = *(const float4*)(ap + i * 4);
                } else {
                    if (gc + 0 < K) v.x = ap[i * 4 + 0];
                    if (gc + 1 < K) v.y = ap[i * 4 + 1];
                    if (gc + 2 < K) v.z = ap[i * 4 + 2];
                    if (gc + 3 < K) v.w = ap[i * 4 + 3];
                }
            }
            aReg[i] = v;
        }
#pragma unroll
        for (int i = 0; i < 2; ++i) {
            const int idx = tid + i * 256;        // float4 id in 32x64 tile
            const int br  = idx >> 4;             // k row 0..31
            const int bc  = (idx & 15) * 4;       // col 0..60
            float4 v = {0.f, 0.f, 0.f, 0.f};
            if (k0 + br < K)
                v = *(const float4*)(B + (k0 + br) * ldb + nBase + bc);
            bReg[i] = v;
        }
    };

    auto storeChunk = [&](int buf) {
        float* __restrict__ as = &As[buf][aRow * ALD_ + aCol];
#pragma unroll
        for (int i = 0; i < 4; ++i)
            *(float4*)(as + i * 4) = aReg[i];
#pragma unroll
        for (int i = 0; i < 2; ++i) {
            const int idx = tid + i * 256;
            const int br  = idx >> 4;
            const int bc  = (idx & 15) * 4;
            float* __restrict__ bs = &Bs[buf][0];
            bs[(bc + 0) * BLD_ + br] = bReg[i].x;  // transpose into [col][k]
            bs[(bc + 1) * BLD_ + br] = bReg[i].y;
            bs[(bc + 2) * BLD_ + br] = bReg[i].z;
            bs[(bc + 3) * BLD_ + br] = bReg[i].w;
        }
    };

    const int nChunk = (K + KC_ - 1) / KC_;

    loadChunk(0);
    storeChunk(0);
    __syncthreads();

    for (int ch = 0; ch < nChunk; ++ch) {
        if (ch + 1 < nChunk) loadChunk((ch + 1) * KC_);

        const int buf = ch & 1;
        const float* __restrict__ as = &As[buf][(wave * 16 + mRow) * ALD_ + kHalf];
        const float* __restrict__ bs = &Bs[buf][nCol * BLD_ + kHalf];
#pragma unroll
        for (int kk = 0; kk < KC_; kk += 4) {
            v2f a = *(const v2f*)(as + kk);
#pragma unroll
            for (int j = 0; j < 4; ++j) {
                v2f b = *(const v2f*)(bs + j * 16 * BLD_ + kk);
                c[j] = __builtin_amdgcn_wmma_f32_16x16x4_f32(
                    false, a, false, b, (short)0, c[j], false, false);
            }
        }

        if (ch + 1 < nChunk) storeChunk((ch + 1) & 1);
        __syncthreads();
    }

    if (rowValid) {
#pragma unroll
        for (int j = 0; j < 4; ++j)
#pragma unroll
            for (int r = 0; r < 8; ++r)
                C[(mTile * 16 + mHalf + r) * ldc + nBase + j * 16 + nCol] = c[j][r];
    }
}

// ---------------------------------------------------------------------------
// Graph preprocessing
// ---------------------------------------------------------------------------
__global__ __launch_bounds__(256) void edge_degree(
    const long long* __restrict__ ei, float* __restrict__ deg, int nEdge)
{
    int e = blockIdx.x * 256 + threadIdx.x;
    if (e >= nEdge) return;
    long long s = ei[e], d = ei[nEdge + e];
    if (s != d) unsafeAtomicAdd(&deg[(int)s], 1.0f);
}

__global__ __launch_bounds__(256) void degree_rsqrt(float* __restrict__ deg, int n)
{
    int i = blockIdx.x * 256 + threadIdx.x;
    if (i >= n) return;
    float d = deg[i];
    deg[i] = (d > 0.0f) ? __frsqrt_rn(d) : 0.0f;
}

__global__ __launch_bounds__(256) void edge_norm(
    const long long* __restrict__ ei, const float* __restrict__ dinv,
    float* __restrict__ norm, int nEdge)
{
    int e = blockIdx.x * 256 + threadIdx.x;
    if (e >= nEdge) return;
    long long s = ei[e], d = ei[nEdge + e];
    norm[e] = (s != d) ? -dinv[(int)s] * dinv[(int)d] : 0.0f;
}

// ---------------------------------------------------------------------------
// Propagation: y[dst] += norm[e] * t[src]  (one wave per edge; lane -> 8 floats)
// ---------------------------------------------------------------------------
__global__ __launch_bounds__(256) void prop_scatter(
    const long long* __restrict__ ei, const float* __restrict__ norm,
    const float* __restrict__ t, float* __restrict__ y, int nEdge)
{
    int e = blockIdx.x * 8 + (threadIdx.x >> 5);      // uniform per wave
    if (e >= nEdge) return;
    float w = norm[e];
    if (w == 0.0f) return;                            // uniform per wave
    int src = (int)ei[e];
    int dst = (int)ei[nEdge + e];
    int lane = threadIdx.x & 31;

    const float4* ts = (const float4*)(t + src * HID_) + lane * 2;
    float4 a0 = ts[0];
    float4 a1 = ts[1];
    float* yd = y + dst * HID_ + lane * 8;
    unsafeAtomicAdd(yd + 0, w * a0.x);
    unsafeAtomicAdd(yd + 1, w * a0.y);
    unsafeAtomicAdd(yd + 2, w * a0.z);
    unsafeAtomicAdd(yd + 3, w * a0.w);
    unsafeAtomicAdd(yd + 4, w * a1.x);
    unsafeAtomicAdd(yd + 5, w * a1.y);
    unsafeAtomicAdd(yd + 6, w * a1.z);
    unsafeAtomicAdd(yd + 7, w * a1.w);
}

// Tx_next = 2*P - Tx_prev   (float4 elementwise; safe when dst == prev)
__global__ __launch_bounds__(256) void cheb_combine(
    const float* __restrict__ P, const float* __restrict__ prev,
    float* __restrict__ dst, int n4)
{
    int i = blockIdx.x * 256 + threadIdx.x;
    if (i >= n4) return;
    float4 p = ((const float4*)P)[i];
    float4 q = ((const float4*)prev)[i];
    float4 r;
    r.x = 2.0f * p.x - q.x;
    r.y = 2.0f * p.y - q.y;
    r.z = 2.0f * p.z - q.z;
    r.w = 2.0f * p.w - q.w;
    ((float4*)dst)[i] = r;
}

// ---------------------------------------------------------------------------
// Epilogue: out = LayerNorm(gelu(out + cheb_b)) * gamma + beta  (one block/row)
// ---------------------------------------------------------------------------
__global__ __launch_bounds__(256) void epilogue_ln_gelu(
    float* __restrict__ out, const float* __restrict__ bias,
    const float* __restrict__ gamma, const float* __restrict__ beta)
{
    __shared__ float red[256];
    const int row = blockIdx.x;
    const int c   = threadIdx.x;
    float x = out[row * HID_ + c] + bias[c];
    float g = 0.5f * x * (1.0f + erff(x * 0.70710678118654752f));  // exact gelu

    red[c] = g;
    __syncthreads();
#pragma unroll
    for (int s = 128; s > 0; s >>= 1) {
        if (c < s) red[c] += red[c + s];
        __syncthreads();
    }
    float mu = red[0] * (1.0f / HID_);
    __syncthreads();

    float d = g - mu;
    red[c] = d * d;
    __syncthreads();
#pragma unroll
    for (int s = 128; s > 0; s >>= 1) {
        if (c < s) red[c] += red[c + s];
        __syncthreads();
    }
    float var = red[0] * (1.0f / HID_);
    out[row * HID_ + c] = d * __frsqrt_rn(var + 1e-5f) * gamma[c] + beta[c];
}

// ---------------------------------------------------------------------------
extern "C" void kernel_launch(void* const* d_in, const int* in_sizes, int n_in,
                              void* d_out, int out_size, void* d_ws, size_t ws_size,
                              hipStream_t stream)
{
    const float*     Preg = (const float*)d_in[0];       // [N_REG, HID]
    const float*     S    = (const float*)d_in[1];       // [N_SEG, N_REG]
    const long long* EI   = (const long long*)d_in[2];   // [2, N_EDGE] int64
    const float*     W    = (const float*)d_in[3];       // [5, HID, HID]
    const float*     bch  = (const float*)d_in[4];       // [HID]
    const float*     gam  = (const float*)d_in[5];       // [HID]
    const float*     bet  = (const float*)d_in[6];       // [HID]

    float* out = (float*)d_out;                          // seg_low      [N_SEG,HID]
    float* raw = out + (size_t)N_SEG_ * HID_;            // seg_low_raw  == Tx0

    const size_t FEAT = (size_t)N_SEG_ * HID_;
    float* ws   = (float*)d_ws;
    float* deg  = ws;                      // N_SEG
    float* nrm  = deg + N_SEG_;            // N_EDGE
    float* bufA = nrm + N_EDGE_;           // Tx1 / Tx3
    float* bufB = bufA + FEAT;             // Tx2 / Tx4
    float* Pp   = bufB + FEAT;             // prop output

    dim3 gGemm((N_SEG_ + BM_ - 1) / BM_, HID_ / BN_);    // (313, 4)
    dim3 gEdgeT((N_EDGE_ + 255) / 256);
    dim3 gEdgeW((N_EDGE_ + 7) / 8);
    dim3 gSeg((N_SEG_ + 255) / 256);
    dim3 gComb((int)(FEAT / 4 + 255) / 256);

    // --- graph normalization ---
    hipMemsetAsync(deg, 0, N_SEG_ * sizeof(float), stream);
    edge_degree<<<gEdgeT, 256, 0, stream>>>(EI, deg, N_EDGE_);
    degree_rsqrt<<<gSeg, 256, 0, stream>>>(deg, N_SEG_);
    edge_norm<<<gEdgeT, 256, 0, stream>>>(EI, deg, nrm, N_EDGE_);

    // --- seg_low_raw = S @ Preg  (Tx0) ---
    wmma_gemm_f32<false><<<gGemm, 256, 0, stream>>>(S, Preg, raw,
                                                    N_SEG_, N_REG_, N_REG_, HID_, HID_);

    // --- out = Tx0 @ W0 ---
    wmma_gemm_f32<false><<<gGemm, 256, 0, stream>>>(raw, W + 0 * HID_ * HID_, out,
                                                    N_SEG_, HID_, HID_, HID_, HID_);

    // --- Tx1 = prop(Tx0);  out += Tx1 @ W1 ---
    hipMemsetAsync(bufA, 0, FEAT * sizeof(float), stream);
    prop_scatter<<<gEdgeW, 256, 0, stream>>>(EI, nrm, raw, bufA, N_EDGE_);
    wmma_gemm_f32<true><<<gGemm, 256, 0, stream>>>(bufA, W + 1 * HID_ * HID_, out,
                                                   N_SEG_, HID_, HID_, HID_, HID_);

    // --- k=2: Tx2 = 2*prop(Tx1) - Tx0 -> bufB ---
    hipMemsetAsync(Pp, 0, FEAT * sizeof(float), stream);
    prop_scatter<<<gEdgeW, 256, 0, stream>>>(EI, nrm, bufA, Pp, N_EDGE_);
    cheb_combine<<<gComb, 256, 0, stream>>>(Pp, raw, bufB, (int)(FEAT / 4));
    wmma_gemm_f32<true><<<gGemm, 256, 0, stream>>>(bufB, W + 2 * HID_ * HID_, out,
                                                   N_SEG_, HID_, HID_, HID_, HID_);

    // --- k=3: Tx3 = 2*prop(Tx2) - Tx1 -> bufA (in place) ---
    hipMemsetAsync(Pp, 0, FEAT * sizeof(float), stream);
    prop_scatter<<<gEdgeW, 256, 0, stream>>>(EI, nrm, bufB, Pp, N_EDGE_);
    cheb_combine<<<gComb, 256, 0, stream>>>(Pp, bufA, bufA, (int)(FEAT / 4));
    wmma_gemm_f32<true><<<gGemm, 256, 0, stream>>>(bufA, W + 3 * HID_ * HID_, out,
                                                   N_SEG_, HID_, HID_, HID_, HID_);

    // --- k=4: Tx4 = 2*prop(Tx3) - Tx2 -> bufB (in place) ---
    hipMemsetAsync(Pp, 0, FEAT * sizeof(float), stream);
    prop_scatter<<<gEdgeW, 256, 0, stream>>>(EI, nrm, bufA, Pp, N_EDGE_);
    cheb_combine<<<gComb, 256, 0, stream>>>(Pp, bufB, bufB, (int)(FEAT / 4));
    wmma_gemm_f32<true><<<gGemm, 256, 0, stream>>>(bufB, W + 4 * HID_ * HID_, out,
                                                   N_SEG_, HID_, HID_, HID_, HID_);

    // --- out = LN(gelu(out + b)) ---
    epilogue_ln_gelu<<<N_SEG_, 256, 0, stream>>>(out, bch, gam, bet);
}